// Covariance_47493748359314
// MI455X (gfx1250) — compile-verified
//
#include <hip/hip_runtime.h>
#include <hip/hip_bf16.h>

typedef __attribute__((ext_vector_type(2))) float v2f;
typedef __attribute__((ext_vector_type(8))) float v8f;
typedef int v4i __attribute__((vector_size(4 * sizeof(int))));

#define GLOBAL_AS __attribute__((address_space(1)))
#define LDS_AS    __attribute__((address_space(3)))

#define B_      64
#define N_      4096
#define D_      256
#define TRI_    (D_ * (D_ + 1) / 2)   // 32896
#define TB2_    4                     // 256/64 : 64-wide blocks per dim
#define NPAIR2_ 10                    // TB2_*(TB2_+1)/2 upper block pairs
#define SLAB_   32                    // K-slab rows staged per iteration
#define NSLAB_  (N_ / SLAB_)          // 128
#define LSTRIDE 136                   // 128 cols + 8 pad floats (bank decorrelation)

#if defined(__has_builtin)
#if __has_builtin(__builtin_amdgcn_global_load_async_to_lds_b128)
#define USE_ASYNC_COPY 1
#endif
#if __has_builtin(__builtin_amdgcn_s_wait_asynccnt)
#define WAIT_ASYNCCNT(n) __builtin_amdgcn_s_wait_asynccnt(n)
#endif
#endif
#ifndef USE_ASYNC_COPY
#define USE_ASYNC_COPY 0
#endif
#ifndef WAIT_ASYNCCNT
#define WAIT_ASYNCCNT(n) asm volatile("s_wait_asynccnt %0" ::"i"(n) : "memory")
#endif

// ---------------- Kernel 1: column means ----------------
__global__ __launch_bounds__(256) void cov_colmean_kernel(
    const float* __restrict__ X, float* __restrict__ means)
{
    const int b = blockIdx.x;
    const int d = threadIdx.x;
    const float* p = X + (size_t)b * (N_ * D_) + d;
    float s = 0.0f;
#pragma unroll 8
    for (int n = 0; n < N_; ++n) {
        s += p[(size_t)n * D_];
    }
    means[b * D_ + d] = s * (1.0f / (float)N_);
}

// ---------------- Kernel 2: LDS-staged Gram via WMMA f32 ----------------
// Block (4 waves) computes a 64x64 region of G = X^T X. A 32x128 K-slab
// (A-panel cols + B-panel cols) is async-copied into double-buffered LDS;
// each wave computes one 32x32 quadrant (2x2 v_wmma_f32_16x16x4_f32 tiles).
__global__ __launch_bounds__(128) void cov_wmma_kernel(
    const float* __restrict__ X,
    const float* __restrict__ means,
    float* __restrict__ out)
{
    __shared__ __align__(16) float lds[2][SLAB_][LSTRIDE];

    const int wave = threadIdx.x >> 5;
    const int lane = threadIdx.x & 31;

    const int b = blockIdx.x / NPAIR2_;
    int p       = blockIdx.x - b * NPAIR2_;
    int bi = 0;
    while (p >= TB2_ - bi) { p -= TB2_ - bi; ++bi; }
    const int bj = bi + p;                       // 64-block pair, bi <= bj

    const int wi = wave >> 1;                    // quadrant row (0/1)
    const int wj = wave & 1;                     // quadrant col (0/1)
    const int m    = lane & 15;
    const int half = lane >> 4;

    const float* Xb = X + (size_t)b * (N_ * D_);
    const int sA = bi * 64;
    const int sB = bj * 64;

    // slab copy: lanes 0-15 fill A-panel (LDS cols 0..63), 16-31 fill B-panel
    const int gcol = (lane < 16) ? (sA + lane * 4) : (sB + (lane - 16) * 4);
    const int lcol = lane * 4;

    // fragment columns inside the LDS slab
    const int cA0 = wi * 32 + m;
    const int cA1 = cA0 + 16;
    const int cB0 = 64 + wj * 32 + m;
    const int cB1 = cB0 + 16;

    v8f acc00 = {}; v8f acc01 = {}; v8f acc10 = {}; v8f acc11 = {};

    auto issue = [&](int slab) {
        float* buf = &lds[slab & 1][0][0];
#pragma unroll
        for (int k = 0; k < 8; ++k) {
            const int r = wave + 4 * k;                       // 8 rows per wave
            const float* src = Xb + (size_t)(slab * SLAB_ + r) * D_ + gcol;
            float* dst = buf + r * LSTRIDE + lcol;
#if USE_ASYNC_COPY
            __builtin_amdgcn_global_load_async_to_lds_b128(
                (GLOBAL_AS v4i*)src, (LDS_AS v4i*)dst, 0, 0);
#else
            *(float4*)dst = *(const float4*)src;
#endif
        }
    };

    issue(0);
    issue(1);

    for (int s = 0; s < NSLAB_; ++s) {
#if USE_ASYNC_COPY
        if (s + 1 < NSLAB_) WAIT_ASYNCCNT(8);   // retire slab s, keep s+1 in flight
        else                WAIT_ASYNCCNT(0);
#endif
        __syncthreads();

        const float* buf = &lds[s & 1][0][0];
#pragma unroll
        for (int kk = 0; kk < SLAB_; kk += 4) {
            const float* r0 = buf + (kk + 2 * half) * LSTRIDE;
            const float* r1 = r0 + LSTRIDE;
            v2f A0 = { r0[cA0], r1[cA0] };
            v2f A1 = { r0[cA1], r1[cA1] };
            v2f B0 = { r0[cB0], r1[cB0] };
            v2f B1 = { r0[cB1], r1[cB1] };
            acc00 = __builtin_amdgcn_wmma_f32_16x16x4_f32(
                false, A0, false, B0, (short)0, acc00, false, false);
            acc01 = __builtin_amdgcn_wmma_f32_16x16x4_f32(
                false, A0, false, B1, (short)0, acc01, false, false);
            acc10 = __builtin_amdgcn_wmma_f32_16x16x4_f32(
                false, A1, false, B0, (short)0, acc10, false, false);
            acc11 = __builtin_amdgcn_wmma_f32_16x16x4_f32(
                false, A1, false, B1, (short)0, acc11, false, false);
        }

        __syncthreads();                         // all waves done reading buf
        if (s + 2 < NSLAB_) issue(s + 2);        // refill the buffer just freed
    }

    // ------------- epilogue: cov = G/N - mu_i*mu_j, packed triu store -------------
    const float invN = 1.0f / (float)N_;
    const float* mb  = means + b * D_;
    const float meanJ0 = mb[bj * 64 + wj * 32 + m];
    const float meanJ1 = mb[bj * 64 + wj * 32 + 16 + m];
    const size_t outbase = (size_t)b * TRI_;

#define COV_EPILOGUE(ACC, SI, SJ)                                          \
    do {                                                                   \
        const float mJ = (SJ) ? meanJ1 : meanJ0;                           \
        const int   J  = bj * 64 + wj * 32 + (SJ) * 16 + m;                \
        _Pragma("unroll")                                                  \
        for (int r = 0; r < 8; ++r) {                                      \
            const int I = bi * 64 + wi * 32 + (SI) * 16 + half * 8 + r;    \
            if (J >= I) {                                                  \
                const float mI  = mb[I];                                   \
                const int   off = I * D_ - (I * (I - 1)) / 2 + (J - I);    \
                out[outbase + off] = ACC[r] * invN - mI * mJ;              \
            }                                                              \
        }                                                                  \
    } while (0)

    COV_EPILOGUE(acc00, 0, 0);
    COV_EPILOGUE(acc01, 0, 1);
    COV_EPILOGUE(acc10, 1, 0);
    COV_EPILOGUE(acc11, 1, 1);
#undef COV_EPILOGUE
}

extern "C" void kernel_launch(void* const* d_in, const int* in_sizes, int n_in,
                              void* d_out, int out_size, void* d_ws, size_t ws_size,
                              hipStream_t stream)
{
    const float* X     = (const float*)d_in[0];
    float*       out   = (float*)d_out;
    float*       means = (float*)d_ws;   // 64*256 floats = 64 KB scratch

    cov_colmean_kernel<<<B_, D_, 0, stream>>>(X, means);

    // 64 batches * 10 upper 64-block pairs = 640 blocks, 4 waves each
    cov_wmma_kernel<<<B_ * NPAIR2_, 128, 0, stream>>>(X, means, out);
}